// NIQE_38886633898366
// MI455X (gfx1250) — compile-verified
//
#include <hip/hip_runtime.h>
#include <hip/hip_bf16.h>
#include <math.h>

#define BATCH 4
#define H1 1056
#define W1 1920
#define H2 528
#define W2 960
#define NB 220      // blocks per image per scale (11 x 20)
#define NBW 20
#define NGAM 9801

typedef float v2f __attribute__((ext_vector_type(2)));
typedef float v8f __attribute__((ext_vector_type(8)));

// ---------------- GAM / R_GAM tables (device lgammaf) ----------------
__global__ void k_tables(float* __restrict__ gam, float* __restrict__ rgam) {
    int i = blockIdx.x * blockDim.x + threadIdx.x;
    if (i >= NGAM) return;
    float g = 0.2f + 0.001f * (float)i;
    gam[i] = g;
    rgam[i] = expf(2.0f * lgammaf(2.0f / g) - lgammaf(1.0f / g) - lgammaf(3.0f / g));
}

// ---------------- grayscale + round (crop 1080 -> 1056 rows) ----------------
__global__ void k_gray(const float* __restrict__ x, float* __restrict__ y) {
    long i = (long)blockIdx.x * blockDim.x + threadIdx.x;
    const long n = (long)BATCH * H1 * W1;
    if (i >= n) return;
    int w = (int)(i % W1);
    long t = i / W1;
    int h = (int)(t % H1);
    int b = (int)(t / H1);
    const long HW = 1080L * 1920L;
    long base = (long)b * 3 * HW + (long)h * 1920 + w;
    float r = x[base], g = x[base + HW], bl = x[base + 2 * HW];
    y[i] = rintf((0.299f * r + 0.587f * g + 0.114f * bl) * 255.0f); // round-half-even
}

// ---------------- separable 7-tap Gaussian (sigma = 7/6), replicate edges ---
__device__ __forceinline__ void gauss7(float* gw) {
    const float s = 7.0f / 6.0f;
    float sum = 0.0f;
#pragma unroll
    for (int i = 0; i < 7; ++i) { float d = (float)(i - 3); gw[i] = expf(-d * d / (2.0f * s * s)); sum += gw[i]; }
#pragma unroll
    for (int i = 0; i < 7; ++i) gw[i] /= sum;
}

__global__ void k_hblur(const float* __restrict__ y, float* __restrict__ hmu,
                        float* __restrict__ hm2, int H, int W) {
    long i = (long)blockIdx.x * blockDim.x + threadIdx.x;
    const long n = (long)BATCH * H * W;
    if (i >= n) return;
    int w = (int)(i % W);
    long t = i / W;
    int h = (int)(t % H);
    int b = (int)(t / H);
    const float* row = y + ((long)b * H + h) * W;
    __builtin_prefetch(row + ((w + 256 < W) ? w + 256 : W - 1), 0, 1); // global_prefetch_b8
    float gw[7]; gauss7(gw);
    float s1 = 0.0f, s2 = 0.0f;
#pragma unroll
    for (int d = -3; d <= 3; ++d) {
        int c = w + d; c = c < 0 ? 0 : (c >= W ? W - 1 : c);
        float v = row[c];
        s1 += gw[d + 3] * v;
        s2 += gw[d + 3] * v * v;
    }
    hmu[i] = s1; hm2[i] = s2;
}

__global__ void k_vnorm(const float* __restrict__ y, const float* __restrict__ hmu,
                        const float* __restrict__ hm2, float* __restrict__ norm,
                        int H, int W) {
    long i = (long)blockIdx.x * blockDim.x + threadIdx.x;
    const long n = (long)BATCH * H * W;
    if (i >= n) return;
    int w = (int)(i % W);
    long t = i / W;
    int h = (int)(t % H);
    int b = (int)(t / H);
    float gw[7]; gauss7(gw);
    float mu = 0.0f, m2 = 0.0f;
#pragma unroll
    for (int d = -3; d <= 3; ++d) {
        int r = h + d; r = r < 0 ? 0 : (r >= H ? H - 1 : r);
        long idx = ((long)b * H + r) * W + w;
        mu += gw[d + 3] * hmu[idx];
        m2 += gw[d + 3] * hm2[idx];
    }
    float sigma = sqrtf(fabsf(m2 - mu * mu));
    norm[i] = (y[i] - mu) / (sigma + 1.0f);
}

// ---------------- bicubic half resize (two separable passes) ----------------
__device__ __forceinline__ float cubicw(float x) {
    float ax = fabsf(x), ax2 = ax * ax, ax3 = ax2 * ax;
    float r = 0.0f;
    if (ax <= 1.0f)      r = 1.5f * ax3 - 2.5f * ax2 + 1.0f;
    else if (ax <= 2.0f) r = -0.5f * ax3 + 2.5f * ax2 - 4.0f * ax + 2.0f;
    return r;
}

__global__ void k_resize_h(const float* __restrict__ y, float* __restrict__ out) {
    // in: B x H1 x W1 ; out: B x H2 x W1
    long i = (long)blockIdx.x * blockDim.x + threadIdx.x;
    const long n = (long)BATCH * H2 * W1;
    if (i >= n) return;
    int w = (int)(i % W1);
    long t = i / W1;
    int oy = (int)(t % H2);
    int b = (int)(t / H2);
    float u = 2.0f * oy + 1.5f;     // scale=0.5 -> u = 2*(o+1) - 0.5
    int left = 2 * oy - 3;          // floor(u - 4)
    float ws[10], sum = 0.0f, acc = 0.0f;
#pragma unroll
    for (int p = 0; p < 10; ++p) { ws[p] = 0.5f * cubicw(0.5f * (u - (float)(left + p))); sum += ws[p]; }
#pragma unroll
    for (int p = 0; p < 10; ++p) {
        int i0 = left + p - 1;
        if (i0 < 0) i0 = -i0 - 1;
        if (i0 >= H1) i0 = 2 * H1 - i0 - 1;
        acc += ws[p] * y[((long)b * H1 + i0) * W1 + w];
    }
    out[i] = acc / sum;
}

__global__ void k_resize_w(const float* __restrict__ y, float* __restrict__ out) {
    // in: B x H2 x W1 ; out: B x H2 x W2
    long i = (long)blockIdx.x * blockDim.x + threadIdx.x;
    const long n = (long)BATCH * H2 * W2;
    if (i >= n) return;
    int ox = (int)(i % W2);
    long t = i / W2;
    int h = (int)(t % H2);
    int b = (int)(t / H2);
    float u = 2.0f * ox + 1.5f;
    int left = 2 * ox - 3;
    float ws[10], sum = 0.0f, acc = 0.0f;
#pragma unroll
    for (int p = 0; p < 10; ++p) { ws[p] = 0.5f * cubicw(0.5f * (u - (float)(left + p))); sum += ws[p]; }
    const float* row = y + ((long)b * H2 + h) * W1;
#pragma unroll
    for (int p = 0; p < 10; ++p) {
        int i0 = left + p - 1;
        if (i0 < 0) i0 = -i0 - 1;
        if (i0 >= W1) i0 = 2 * W1 - i0 - 1;
        acc += ws[p] * row[i0];
    }
    out[i] = acc / sum;
}

// ---------------- per-block AGGD features (LDS-tiled) ----------------
__device__ __forceinline__ float bred_sum(float v, float* red, int tid) {
    red[tid] = v; __syncthreads();
    for (int s = 128; s > 0; s >>= 1) { if (tid < s) red[tid] += red[tid + s]; __syncthreads(); }
    float r = red[0]; __syncthreads();
    return r;
}

__global__ void k_feats(const float* __restrict__ norm, const float* __restrict__ gam,
                        const float* __restrict__ rgam, float* __restrict__ dist,
                        int H, int W, int bs, int featoff) {
    __shared__ float tile[96 * 96];     // 36 KB of the 320 KB WGP LDS
    __shared__ float red[256];
    __shared__ int   redi[256];
    int tid = threadIdx.x;
    int blk = blockIdx.x;
    int b  = blk / NB, r0 = blk % NB;
    int ih = r0 / NBW, iw = r0 % NBW;
    const float* src = norm + ((long)b * H + (long)ih * bs) * W + (long)iw * bs;
    int N = bs * bs;
    for (int e = tid; e < N; e += 256) {
        int rr = e / bs, cc = e % bs;
        tile[e] = src[(long)rr * W + cc];
    }
    __syncthreads();

    const int sh0[5] = {0, 0, 1, 1, 1};
    const int sh1[5] = {0, 1, 0, 1, -1};
    float feats[18];

    for (int s = 0; s < 5; ++s) {
        float ssn = 0.f, ssp = 0.f, cn = 0.f, cp = 0.f, sab = 0.f, ssq = 0.f;
        for (int e = tid; e < N; e += 256) {
            int rr = e / bs, cc = e % bs;
            float v = tile[e];
            if (s > 0) {
                int r2 = rr - sh0[s]; if (r2 < 0) r2 += bs;
                int c2 = cc - sh1[s]; if (c2 < 0) c2 += bs; if (c2 >= bs) c2 -= bs;
                v = v * tile[r2 * bs + c2];
            }
            if (v < 0.f)      { ssn += v * v; cn += 1.f; }
            else if (v > 0.f) { ssp += v * v; cp += 1.f; }
            sab += fabsf(v);
            ssq += v * v;
        }
        ssn = bred_sum(ssn, red, tid);
        ssp = bred_sum(ssp, red, tid);
        cn  = bred_sum(cn,  red, tid);
        cp  = bred_sum(cp,  red, tid);
        sab = bred_sum(sab, red, tid);
        ssq = bred_sum(ssq, red, tid);

        float left_std  = sqrtf(ssn / cn);
        float right_std = sqrtf(ssp / cp);
        float gh  = left_std / right_std;
        float fN  = (float)N;
        float rhat = (sab / fN) * (sab / fN) / (ssq / fN);
        float g2 = gh * gh;
        float rhatnorm = rhat * (gh * g2 + 1.f) * (gh + 1.f) / ((g2 + 1.f) * (g2 + 1.f));

        // argmin over R_GAM table
        float best = 3.4e38f; int bi = 0;
        for (int i = tid; i < NGAM; i += 256) {
            float d = rgam[i] - rhatnorm; d = d * d;
            if (d < best) { best = d; bi = i; }
        }
        red[tid] = best; redi[tid] = bi; __syncthreads();
        for (int st = 128; st > 0; st >>= 1) {
            if (tid < st) {
                if (red[tid + st] < red[tid] ||
                    (red[tid + st] == red[tid] && redi[tid + st] < redi[tid])) {
                    red[tid] = red[tid + st]; redi[tid] = redi[tid + st];
                }
            }
            __syncthreads();
        }
        int idx = redi[0]; __syncthreads();

        float alpha = gam[idx];
        float ratio = expf(0.5f * (lgammaf(1.0f / alpha) - lgammaf(3.0f / alpha)));
        float bl = left_std * ratio, br = right_std * ratio;
        if (s == 0) { feats[0] = alpha; feats[1] = 0.5f * (bl + br); }
        else {
            int o = 2 + (s - 1) * 4;
            feats[o]     = alpha;
            feats[o + 1] = (br - bl) * expf(lgammaf(2.0f / alpha) - lgammaf(1.0f / alpha));
            feats[o + 2] = bl;
            feats[o + 3] = br;
        }
    }
    if (tid == 0) {
        float* dst = dist + ((long)b * NB + r0) * 36 + featoff;
#pragma unroll
        for (int f = 0; f < 18; ++f) dst[f] = feats[f];
    }
}

// ---------------- final: mu, WMMA covariance, GJ inverse, quad form ----------
__global__ void k_final(const float* __restrict__ dist, const float* __restrict__ mu_pris,
                        const float* __restrict__ cov_pris, float* __restrict__ out) {
    __shared__ float smem[224 * 48];   // phase 1: xc (224x48); phase 2: cov(48x48) @0, aug(36x72) @4096
    __shared__ float mu[36];
    __shared__ float fac[36];
    __shared__ float sc[1];
    __shared__ float red[288];

    int tid = threadIdx.x;
    int b = blockIdx.x;
    const float* D = dist + (long)b * NB * 36;

    if (tid < 36) {
        float s = 0.f;
        for (int n = 0; n < NB; ++n) s += D[n * 36 + tid];
        mu[tid] = s / (float)NB;
    }
    __syncthreads();

    // xc padded to 224 x 48 in LDS
    for (int e = tid; e < 224 * 48; e += 288) {
        int n = e / 48, f = e % 48;
        smem[e] = (n < NB && f < 36) ? (D[n * 36 + f] - mu[f]) : 0.f;
    }
    __syncthreads();

    // cov = xc^T * xc via V_WMMA_F32_16X16X4_F32; 9 waves, one 16x16 tile each.
    // Branch-free operand gather: per-lane LDS offsets precomputed so EXEC stays
    // all-ones through the whole K-loop (no saveexec around the ds_loads).
    int wave = tid >> 5, lane = tid & 31;
    int half = lane >> 4;          // 0: lanes 0-15, 1: lanes 16-31
    int l15  = lane & 15;
    int mt = (wave / 3) * 16, nt = (wave % 3) * 16;
    // A (16x4, M=f): lanes 0-15 hold K0/K1, lanes 16-31 hold K2/K3
    int aoff0 = (2 * half) * 48 + mt + l15;      // K = 2*half
    int aoff1 = aoff0 + 48;                      // K = 2*half + 1
    // B (4x16, N=g): VGPR0 = rows K0(K1 upper), VGPR1 = rows K2(K3 upper)
    int boff0 = half * 48 + nt + l15;            // K = half
    int boff1 = boff0 + 96;                      // K = half + 2
    v8f c = {0.f, 0.f, 0.f, 0.f, 0.f, 0.f, 0.f, 0.f};
    for (int k0 = 0; k0 < 224; k0 += 4) {
        int kb = k0 * 48;
        v2f a, bb;
        a.x  = smem[kb + aoff0];
        a.y  = smem[kb + aoff1];
        bb.x = smem[kb + boff0];
        bb.y = smem[kb + boff1];
        c = __builtin_amdgcn_wmma_f32_16x16x4_f32(false, a, false, bb,
                                                  (short)0, c, false, false);
    }
    __syncthreads();   // all waves done reading xc; safe to repurpose smem

    // C layout: VGPR j -> M = j (+8 for upper lanes), N = lane[3:0]
#pragma unroll
    for (int j = 0; j < 8; ++j) {
        int row = mt + j + half * 8;
        int col = nt + l15;
        smem[row * 48 + col] = c[j];
    }
    __syncthreads();

    // aug = [ (cov_pris + cov/219)/2 | I ]  (36 x 72)
    float* aug = smem + 4096;
    for (int e = tid; e < 36 * 72; e += 288) {
        int r = e / 72, cc = e % 72;
        float v;
        if (cc < 36) v = 0.5f * (cov_pris[((long)b * 36 + r) * 36 + cc] +
                                 smem[r * 48 + cc] * (1.0f / 219.0f));
        else         v = ((cc - 36) == r) ? 1.f : 0.f;
        aug[e] = v;
    }

    // Gauss-Jordan inverse (matrix is SPD-ish: cov_pris has +1e-3 I)
    for (int k = 0; k < 36; ++k) {
        __syncthreads();
        if (tid == 0) sc[0] = 1.0f / aug[k * 72 + k];
        __syncthreads();
        float ip = sc[0];
        if (tid < 72) aug[k * 72 + tid] *= ip;
        __syncthreads();
        if (tid < 36) fac[tid] = (tid == k) ? 0.f : aug[tid * 72 + k];
        __syncthreads();
        for (int e = tid; e < 36 * 72; e += 288) {
            int r = e / 72, cc = e % 72;
            if (r != k) aug[e] -= fac[r] * aug[k * 72 + cc];
        }
    }
    __syncthreads();

    // q = diff^T * inv * diff ; out = sqrt(q)
    float q = 0.f;
    if (tid < 36) {
        float df = mu_pris[b * 36 + tid] - mu[tid];
        float srow = 0.f;
        for (int g = 0; g < 36; ++g) {
            float dg = mu_pris[b * 36 + g] - mu[g];
            srow += aug[tid * 72 + 36 + g] * dg;
        }
        q = df * srow;
    }
    red[tid] = q; __syncthreads();
    if (tid == 0) {
        float s = 0.f;
        for (int i = 0; i < 36; ++i) s += red[i];
        out[b] = sqrtf(fmaxf(s, 0.f));
    }
}

// ---------------- launch ----------------
extern "C" void kernel_launch(void* const* d_in, const int* in_sizes, int n_in,
                              void* d_out, int out_size, void* d_ws, size_t ws_size,
                              hipStream_t stream) {
    const float* x        = (const float*)d_in[0];
    const float* mu_pris  = (const float*)d_in[1];
    const float* cov_pris = (const float*)d_in[2];
    float* out = (float*)d_out;
    float* ws  = (float*)d_ws;

    const size_t NY  = (size_t)BATCH * H1 * W1;   // 8,110,080
    const size_t NY2 = (size_t)BATCH * H2 * W2;   // 2,027,520
    const size_t NRT = (size_t)BATCH * H2 * W1;   // 4,055,040

    float* y1    = ws;
    float* norm1 = ws + NY;
    float* bufA  = ws + 2 * NY;
    float* bufB  = ws + 3 * NY;
    float* rt    = bufA;                // reuse: free between vnorm(1) and hblur(2)
    float* y2    = ws + 4 * NY;
    float* norm2 = y2 + NY2;
    float* dist  = norm2 + NY2;
    float* gam   = dist + (size_t)BATCH * NB * 36;
    float* rgam  = gam + NGAM;

    k_tables<<<(NGAM + 255) / 256, 256, 0, stream>>>(gam, rgam);
    k_gray<<<(unsigned)((NY + 255) / 256), 256, 0, stream>>>(x, y1);

    // scale 1
    k_hblur<<<(unsigned)((NY + 255) / 256), 256, 0, stream>>>(y1, bufA, bufB, H1, W1);
    k_vnorm<<<(unsigned)((NY + 255) / 256), 256, 0, stream>>>(y1, bufA, bufB, norm1, H1, W1);
    k_feats<<<BATCH * NB, 256, 0, stream>>>(norm1, gam, rgam, dist, H1, W1, 96, 0);

    // half resize
    k_resize_h<<<(unsigned)((NRT + 255) / 256), 256, 0, stream>>>(y1, rt);
    k_resize_w<<<(unsigned)((NY2 + 255) / 256), 256, 0, stream>>>(rt, y2);

    // scale 2
    k_hblur<<<(unsigned)((NY2 + 255) / 256), 256, 0, stream>>>(y2, bufA, bufB, H2, W2);
    k_vnorm<<<(unsigned)((NY2 + 255) / 256), 256, 0, stream>>>(y2, bufA, bufB, norm2, H2, W2);
    k_feats<<<BATCH * NB, 256, 0, stream>>>(norm2, gam, rgam, dist, H2, W2, 48, 18);

    // per-batch: mu, WMMA covariance, inverse, quadratic form
    k_final<<<BATCH, 288, 0, stream>>>(dist, mu_pris, cov_pris, out);
}